// BondCenteredTensorMomentDescriptor_14242111553895
// MI455X (gfx1250) — compile-verified
//
#include <hip/hip_runtime.h>
#include <math.h>

// ============================================================================
// Bond-centered tensor-moment descriptor for MI455X (gfx1250).
// Reformulated as per-feature GEMMs over edge tiles, executed with the CDNA5
// fp32 matrix instruction V_WMMA_F32_16X16X4_F32 (wave32).
//
// Pipeline (all on `stream`, graph-capture safe, deterministic):
//   1. clear_kernel    : zero fused-coefficient matrices in workspace
//   2. build_cg_kernel : real Clebsch-Gordan blocks (fp64, mirrors reference)
//   3. build_A_kernel  : fuse CG with w1 -> A1[f][84][32], w2 -> A2[f][228][32]
//   4. tp_edge_kernel  : main WMMA kernel (156 wmma per 16-edge tile per f)
// ============================================================================

typedef float v2f __attribute__((ext_vector_type(2)));
typedef float v8f __attribute__((ext_vector_type(8)));

#define CUTOFF_F 5.0f
#define GAMMA_F  5.0f

// ------------------------- compile-time sizing ------------------------------
__host__ __device__ constexpr int c_abs(int x){ return x < 0 ? -x : x; }
__host__ __device__ constexpr int c_min(int a,int b){ return a < b ? a : b; }

__host__ __device__ constexpr int n_paths(int L1,int L2,int L3){
  int n = 0;
  for (int l1 = 0; l1 <= L1; ++l1)
    for (int l2 = 0; l2 <= L2; ++l2)
      for (int l3 = c_abs(l1-l2); l3 <= c_min(l1+l2, L3); ++l3) ++n;
  return n;
}
constexpr int NP1 = n_paths(2,2,4);   // 19 paths, matches w1 rows
constexpr int NP2 = n_paths(4,2,4);   // 33 paths, matches w2 rows

// CG blocks stored in paths(4,2,4) triple order (superset of paths(2,2,4))
__host__ __device__ constexpr int cg_total(){
  int t = 0;
  for (int l1 = 0; l1 <= 4; ++l1)
    for (int l2 = 0; l2 <= 2; ++l2)
      for (int l3 = c_abs(l1-l2); l3 <= c_min(l1+l2,4); ++l3)
        t += (2*l1+1)*(2*l2+1)*(2*l3+1);
  return t;
}
constexpr int CG_TOTAL = cg_total();  // ~4013 floats

constexpr int K1   = 84;    // 81 (9x9 pair space) padded to K-step multiple of 4
constexpr int K2   = 228;   // 225 (25x9 pair space) padded
constexpr int NCOL = 32;    // 25 outputs padded to two 16-wide WMMA N tiles
constexpr int A1_OFF = CG_TOTAL;
constexpr int A1_SZ  = 32 * K1 * NCOL;
constexpr int A2_OFF = A1_OFF + A1_SZ;
constexpr int A2_SZ  = 32 * K2 * NCOL;

// ------------------------- device helpers -----------------------------------
__device__ inline int cg_offset(int L1,int L2,int L3){
  int off = 0;
  for (int l1 = 0; l1 <= 4; ++l1)
    for (int l2 = 0; l2 <= 2; ++l2)
      for (int l3 = (l1>l2? l1-l2 : l2-l1); l3 <= (((l1+l2)<4)?(l1+l2):4); ++l3){
        if (l1==L1 && l2==L2 && l3==L3) return off;
        off += (2*l1+1)*(2*l2+1)*(2*l3+1);
      }
  return -1;
}

__device__ inline void nth_path(int L1,int L2,int L3,int p,int& o1,int& o2,int& o3){
  int n = 0;
  for (int l1 = 0; l1 <= L1; ++l1)
    for (int l2 = 0; l2 <= L2; ++l2)
      for (int l3 = (l1>l2? l1-l2 : l2-l1); l3 <= (((l1+l2)<L3)?(l1+l2):L3); ++l3){
        if (n == p){ o1=l1; o2=l2; o3=l3; return; }
        ++n;
      }
  o1 = o2 = o3 = 0;
}

__device__ inline double factd(int n){
  double r = 1.0;
  for (int i = 2; i <= n; ++i) r *= (double)i;
  return r;
}

// complex-basis CG coefficient (exactly mirrors reference _cg_complex)
__device__ double cg_complex(int j1,int m1,int j2,int m2,int j3,int m3){
  if (m1 + m2 != m3) return 0.0;
  int dj = j1 > j2 ? j1 - j2 : j2 - j1;
  if (j3 < dj || j3 > j1 + j2) return 0.0;
  double pref = sqrt((2.0*j3+1.0) * factd(j1+j2-j3) * factd(j1-j2+j3) *
                     factd(-j1+j2+j3) / factd(j1+j2+j3+1));
  pref *= sqrt(factd(j1+m1)*factd(j1-m1)*factd(j2+m2)*factd(j2-m2)*
               factd(j3+m3)*factd(j3-m3));
  double s = 0.0;
  for (int k = 0; k <= j1 + j2 - j3; ++k){
    int d0=k, d1=j1+j2-j3-k, d2=j1-m1-k, d3=j2+m2-k, d4=j3-j2+m1+k, d5=j3-j1-m2+k;
    if (d0<0||d1<0||d2<0||d3<0||d4<0||d5<0) continue;
    double den = factd(d0)*factd(d1)*factd(d2)*factd(d3)*factd(d4)*factd(d5);
    s += ((k & 1) ? -1.0 : 1.0) / den;
  }
  return pref * s;
}

// nnz of row `a` of the complex->real basis matrix U(l) (reference _U_real)
__device__ inline int u_row(int l,int a,int* col,double* vr,double* vi){
  const double s = 0.70710678118654752440;
  int mr = a - l;
  if (mr == 0){ col[0]=l; vr[0]=1.0; vi[0]=0.0; return 1; }
  if (mr > 0){
    col[0]=l+mr; vr[0]=(mr&1)? -s : s; vi[0]=0.0;   // (-1)^m/sqrt2
    col[1]=l-mr; vr[1]=s;              vi[1]=0.0;   // 1/sqrt2
    return 2;
  }
  int m = -mr;
  col[0]=l-m; vr[0]=0.0; vi[0]=s;                   // i/sqrt2
  col[1]=l+m; vr[1]=0.0; vi[1]=(m&1)? s : -s;       // -i*(-1)^m/sqrt2
  return 2;
}

// ------------------------- init kernels -------------------------------------
__global__ void clear_kernel(float* p, int n){
  for (int i = blockIdx.x*blockDim.x + threadIdx.x; i < n; i += gridDim.x*blockDim.x)
    p[i] = 0.0f;
}

// one block per (l1,l2,l3) triple; compute real CG block R.real + R.imag
__global__ void build_cg_kernel(float* __restrict__ cg){
  int l1,l2,l3; nth_path(4,2,4,(int)blockIdx.x,l1,l2,l3);
  int n1 = 2*l1+1, n2 = 2*l2+1, n3 = 2*l3+1;
  int total = n1*n2*n3;
  int off = cg_offset(l1,l2,l3);
  for (int e = threadIdx.x; e < total; e += blockDim.x){
    int a  = e / (n2*n3);
    int c  = (e / n3) % n2;
    int ee = e % n3;
    int cb[2]; double br[2], bi[2]; int nb = u_row(l1,a ,cb,br,bi);
    int cd[2]; double dr[2], di[2]; int nd = u_row(l2,c ,cd,dr,di);
    int cf[2]; double fr[2], fi[2]; int nf = u_row(l3,ee,cf,fr,fi);
    double ar = 0.0, ai = 0.0;
    for (int x = 0; x < nb; ++x)
      for (int y = 0; y < nd; ++y)
        for (int z = 0; z < nf; ++z){
          int m1 = cb[x]-l1, m2 = cd[y]-l2, m3 = cf[z]-l3;
          if (m1 + m2 != m3) continue;
          double cgv = cg_complex(l1,m1,l2,m2,l3,m3);
          if (cgv == 0.0) continue;
          double r1 = br[x]*dr[y] - bi[x]*di[y];     // U1*U2
          double i1 = br[x]*di[y] + bi[x]*dr[y];
          double cr = fr[z], ci = -fi[z];            // conj(U3)
          double r2 = r1*cr - i1*ci;
          double i2 = r1*ci + i1*cr;
          ar += r2 * cgv; ai += i2 * cgv;
        }
    cg[off + e] = (float)(ar + ai);
  }
}

// fuse CG with per-path weights into A[f][Krows][32]; distinct paths hit
// distinct (row,col) cells so plain stores into the zeroed buffer suffice.
__global__ void build_A_kernel(const float* __restrict__ cg,
                               const float* __restrict__ w,
                               float* __restrict__ A, int L1, int Krows){
  int p = blockIdx.x;
  int l1,l2,l3; nth_path(L1,2,4,p,l1,l2,l3);
  int n1 = 2*l1+1, n2 = 2*l2+1, n3 = 2*l3+1;
  int off = cg_offset(l1,l2,l3);
  int total = n1*n2*n3*32;
  for (int idx = threadIdx.x; idx < total; idx += blockDim.x){
    int e = idx >> 5;
    int f = idx & 31;
    int i = e / (n2*n3);
    int j = (e / n3) % n2;
    int k = e % n3;
    int row = (l1*l1 + i) * 9 + (l2*l2 + j);
    int col = l3*l3 + k;
    A[((size_t)f*Krows + row)*NCOL + col] = cg[off + e] * w[p*32 + f];
  }
}

// ------------------------- main WMMA kernel ---------------------------------
// blockIdx.y = feature f, blockIdx.x = group of 16 edge-tiles (4 waves x 4).
__global__ void __launch_bounds__(128)
tp_edge_kernel(const float* __restrict__ desc,   // [NA][9][32]
               const int*   __restrict__ nidx,   // [E][2]
               const float* __restrict__ disp,   // [E][3]
               const float* __restrict__ A1,     // [32][K1][32]
               const float* __restrict__ A2,     // [32][K2][32]
               float* __restrict__ out,          // [E][25][32]
               int E, int ntiles)
{
  __shared__ float sA1[K1][33];        // stride 33: bank-conflict-free
  __shared__ float sA2[K2][33];
  __shared__ float sD1[4][16][9];
  __shared__ float sD2[4][16][9];
  __shared__ float sY [4][16][27];
  __shared__ float sB [4][16][9];

  const int f    = blockIdx.y;
  const int tid  = threadIdx.x;
  const int w    = tid >> 5;
  const int lane = tid & 31;
  const int m    = lane & 15;          // M index (edge within tile / B column)
  const bool hi  = lane >= 16;

  // cooperative load of this feature's fused matrices into LDS
  const float* a1f = A1 + (size_t)f * K1 * NCOL;
  for (int i = tid; i < K1*NCOL; i += 128) sA1[i/NCOL][i%NCOL] = a1f[i];
  const float* a2f = A2 + (size_t)f * K2 * NCOL;
  for (int i = tid; i < K2*NCOL; i += 128) sA2[i/NCOL][i%NCOL] = a2f[i];
  __syncthreads();

  for (int t = 0; t < 4; ++t){
    const int tile = blockIdx.x * 16 + t * 4 + w;
    const bool active = tile < ntiles;          // wave-uniform

    if (active){
      const int e = tile * 16 + m;
      const bool ev = e < E;
      if (!hi){
        if (ev){
          int ia = nidx[2*e + 0];
          #pragma unroll
          for (int q = 0; q < 9; ++q) sD1[w][m][q] = desc[((size_t)ia*9 + q)*32 + f];
          // bond basis: b[j] = Ysph[j] * cut * damp * sinc((f+1)*r/CUTOFF)
          float dx = disp[3*e+0], dy = disp[3*e+1], dz = disp[3*e+2];
          float r  = sqrtf(dx*dx + dy*dy + dz*dz);
          float rs = fmaxf(r, 1e-9f);
          float x = dx/rs, y = dy/rs, z = dz/rs;
          const float s3 = 1.7320508075688772f;
          float Ys[9];
          Ys[0] = 1.0f; Ys[1] = y; Ys[2] = z; Ys[3] = x;
          Ys[4] = s3*x*y; Ys[5] = s3*y*z; Ys[6] = 0.5f*(3.0f*z*z - 1.0f);
          Ys[7] = s3*x*z; Ys[8] = 0.5f*s3*(x*x - y*y);
          float xx  = r / CUTOFF_F;
          float cut = (xx < 1.0f) ? expf(1.0f - 1.0f/fmaxf(1.0f - xx*xx, 1e-9f)) : 0.0f;
          float damp = expf(-GAMMA_F / rs);
          float arg  = (float)(f + 1) * xx * 3.14159265358979323846f;
          float radial = (arg < 1e-6f) ? 1.0f : (sinf(arg) / arg);
          float env = cut * damp * radial;
          #pragma unroll
          for (int q = 0; q < 9; ++q) sB[w][m][q] = Ys[q] * env;
        }
      } else {
        if (ev){
          int ib = nidx[2*e + 1];
          #pragma unroll
          for (int q = 0; q < 9; ++q) sD2[w][m][q] = desc[((size_t)ib*9 + q)*32 + f];
        }
      }
    }
    __syncthreads();

    if (active){
      // ---- TP1: y[16x32] = Z1[16x84] * A1_f[84x32], K-steps of 4 ----------
      v8f c0 = {}; v8f c1 = {};
      #pragma unroll
      for (int s = 0; s < 21; ++s){
        const int k0 = 4*s + (hi ? 2 : 0);          // compile-time after unroll
        v2f a;
        a.x = (k0   < 81) ? sD1[w][m][ k0   /9] * sD2[w][m][ k0   %9] : 0.0f;
        a.y = (k0+1 < 81) ? sD1[w][m][(k0+1)/9] * sD2[w][m][(k0+1)%9] : 0.0f;
        v2f b0, b1;
        b0.x = sA1[k0][m];      b0.y = sA1[k0+1][m];
        b1.x = sA1[k0][m+16];   b1.y = sA1[k0+1][m+16];
        c0 = __builtin_amdgcn_wmma_f32_16x16x4_f32(false, a, false, b0, (short)0, c0, false, false);
        c1 = __builtin_amdgcn_wmma_f32_16x16x4_f32(false, a, false, b1, (short)0, c1, false, false);
      }
      // D tile -> LDS y (rows v / v+8 per ISA C/D layout), keep k < 25
      const int rb = hi ? 8 : 0;
      #pragma unroll
      for (int v = 0; v < 8; ++v){
        sY[w][rb + v][m] = c0[v];
        if (m < 9) sY[w][rb + v][m + 16] = c1[v];
      }
    }
    __syncthreads();

    if (active){
      // ---- TP2: out[16x32] = Z2[16x228] * A2_f[228x32] --------------------
      v8f d0 = {}; v8f d1v = {};
      #pragma unroll
      for (int s = 0; s < 57; ++s){
        const int k0 = 4*s + (hi ? 2 : 0);
        v2f a;
        a.x = (k0   < 225) ? sY[w][m][ k0   /9] * sB[w][m][ k0   %9] : 0.0f;
        a.y = (k0+1 < 225) ? sY[w][m][(k0+1)/9] * sB[w][m][(k0+1)%9] : 0.0f;
        v2f b0, b1;
        b0.x = sA2[k0][m];      b0.y = sA2[k0+1][m];
        b1.x = sA2[k0][m+16];   b1.y = sA2[k0+1][m+16];
        d0  = __builtin_amdgcn_wmma_f32_16x16x4_f32(false, a, false, b0, (short)0, d0,  false, false);
        d1v = __builtin_amdgcn_wmma_f32_16x16x4_f32(false, a, false, b1, (short)0, d1v, false, false);
      }
      const int rb = hi ? 8 : 0;
      #pragma unroll
      for (int v = 0; v < 8; ++v){
        const int e = tile*16 + rb + v;
        if (e < E){
          out[((size_t)e*25 + m)*32 + f] = d0[v];                 // k = 0..15
          if (m < 9) out[((size_t)e*25 + m + 16)*32 + f] = d1v[v]; // k = 16..24
        }
      }
    }
    __syncthreads();
  }
}

// ------------------------- entry point ---------------------------------------
extern "C" void kernel_launch(void* const* d_in, const int* in_sizes, int n_in,
                              void* d_out, int out_size, void* d_ws, size_t ws_size,
                              hipStream_t stream) {
  const float* desc = (const float*)d_in[0];
  const int*   nidx = (const int*)  d_in[1];
  const float* disp = (const float*)d_in[2];
  const float* w1   = (const float*)d_in[3];
  const float* w2   = (const float*)d_in[4];
  float* out = (float*)d_out;
  float* ws  = (float*)d_ws;

  float* cg = ws;
  float* A1 = ws + A1_OFF;
  float* A2 = ws + A2_OFF;

  const int E      = in_sizes[1] / 2;
  const int ntiles = (E + 15) / 16;
  const int gx     = (ntiles + 15) / 16;    // 4 waves x 4 tiles per block

  clear_kernel   <<<256, 256, 0, stream>>>(A1, A1_SZ + A2_SZ);
  build_cg_kernel<<<NP2, 128, 0, stream>>>(cg);
  build_A_kernel <<<NP1, 256, 0, stream>>>(cg, w1, A1, 2, K1);
  build_A_kernel <<<NP2, 256, 0, stream>>>(cg, w2, A2, 4, K2);
  tp_edge_kernel <<<dim3(gx, 32), 128, 0, stream>>>(desc, nidx, disp, A1, A2, out, E, ntiles);
}